// ResidualVectorQuantize_28003186770498
// MI455X (gfx1250) — compile-verified
//
#include <hip/hip_runtime.h>

// ---------------------------------------------------------------------------
// Residual Vector Quantize (eval forward), MI455X / gfx1250, wave32.
//
// Key algebraic reductions vs. the reference:
//  * gate == onehot(argmax probs)  -> only the selected expert's NN search and
//    codeword matter per token.
//  * z_q_total == z - residual_final -> no z_q accumulator; one fused pass.
//  * commitment_loss == codebook_loss (forward values identical).
//
// Perf model: memory-bound at ~3.6 GB of HBM traffic (~155us @ 23.3 TB/s).
// in_proj ([BT,512]x[512,8]) stays VALU (coalesced lane<->token loads of the
// [c,t]-major residual). out_proj ([BT,8]x[8,512]) uses V_WMMA_F32_16X16X4_F32
// (fp32-exact), fused with the residual read-modify-write via an LDS transpose.
// ---------------------------------------------------------------------------

#define NQ  9
#define EE  8
#define KK  128
#define DCB 8
#define DIN 512
#define BB  16
#define TT  4096
#define BT  (BB*TT)

typedef float v2f __attribute__((ext_vector_type(2)));
typedef float v8f __attribute__((ext_vector_type(8)));

// ---------------------------------------------------------------------------
// Prep: weight-norm w_in / w_out, normalized codebooks (+ squared norms),
// zero the loss accumulators (every launch -> graph-replay deterministic).
// ws_acc layout: [0..71] probs-sum[i*8+e], [72..143] onehot-count, [144..152] sq-err
// ---------------------------------------------------------------------------
__global__ void rvq_prep(const float* __restrict__ in_v, const float* __restrict__ in_g,
                         const float* __restrict__ out_v, const float* __restrict__ out_g,
                         const float* __restrict__ cbks,
                         float* __restrict__ ws_win, float* __restrict__ ws_wout,
                         float* __restrict__ ws_cbn, float* __restrict__ ws_cbnsq,
                         float* __restrict__ ws_acc)
{
  const int N_WIN = NQ*DCB;        // 72 rows of 512
  const int N_WOUT = NQ*DIN;       // 4608 rows of 8
  const int N_CBN = NQ*EE*KK;      // 9216 rows of 8
  int tid = blockIdx.x*blockDim.x + threadIdx.x;
  if (tid < N_WIN) {
    const float* v = in_v + (size_t)tid*DIN;
    float s = 0.f;
    for (int c = 0; c < DIN; ++c) s += v[c]*v[c];
    float sc = in_g[tid] / sqrtf(s);
    float* o = ws_win + (size_t)tid*DIN;
    for (int c = 0; c < DIN; ++c) o[c] = v[c]*sc;
  } else if (tid < N_WIN + N_WOUT) {
    int row = tid - N_WIN;
    const float* v = out_v + (size_t)row*DCB;
    float s = 0.f;
    #pragma unroll
    for (int d = 0; d < DCB; ++d) s += v[d]*v[d];
    float sc = out_g[row] / sqrtf(s);
    float* o = ws_wout + (size_t)row*DCB;
    #pragma unroll
    for (int d = 0; d < DCB; ++d) o[d] = v[d]*sc;
  } else if (tid < N_WIN + N_WOUT + N_CBN) {
    int row = tid - N_WIN - N_WOUT;
    const float* v = cbks + (size_t)row*DCB;
    float s = 0.f;
    #pragma unroll
    for (int d = 0; d < DCB; ++d) s += v[d]*v[d];
    float inv = 1.f / fmaxf(sqrtf(s), 1e-12f);
    float* o = ws_cbn + (size_t)row*DCB;
    float s2 = 0.f;
    #pragma unroll
    for (int d = 0; d < DCB; ++d) { float n = v[d]*inv; o[d] = n; s2 += n*n; }
    ws_cbnsq[row] = s2;
  } else if (tid < N_WIN + N_WOUT + N_CBN + 160) {
    ws_acc[tid - (N_WIN + N_WOUT + N_CBN)] = 0.f;
  }
}

// ---------------------------------------------------------------------------
// Stage A (per quantizer): in_proj + router + softmax/argmax + expert NN search.
// One token per lane; 256 tokens per block; grid = BT/256 = 256 blocks.
// Emits: q (selected raw codeword), codes, latents (z_e), loss partials.
// ---------------------------------------------------------------------------
__global__ __launch_bounds__(256) void rvq_stageA(
    const float* __restrict__ resid,   // [B][DIN][T]
    const float* __restrict__ win,     // [DCB][DIN] weight-normed
    const float* __restrict__ inb,     // [DCB]
    const float* __restrict__ w1,      // [DCB][DCB]
    const float* __restrict__ b1,      // [DCB]
    const float* __restrict__ w2,      // [EE][DCB]
    const float* __restrict__ b2,      // [EE]
    const float* __restrict__ cbn,     // [EE][KK][DCB] normalized
    const float* __restrict__ cbnsq,   // [EE][KK]
    const float* __restrict__ cbraw,   // [EE][KK][DCB] raw
    float* __restrict__ qout,          // [BT][DCB]
    float* __restrict__ codes,         // [B][NQ][T] (float values)
    float* __restrict__ lat,           // [B][NQ*T][DCB]
    float* __restrict__ acc,           // ws_acc
    int qi)
{
  __shared__ float s_win[DIN*DCB];            // [c][d]
  __shared__ float s_cbn[EE*(KK*DCB + 8)];    // per-expert pad 8 -> bank spread
  __shared__ float s_csq[EE*KK];
  __shared__ float s_w1[DCB*DCB];
  __shared__ float s_b1[DCB];
  __shared__ float s_w2[EE*DCB];
  __shared__ float s_b2[EE];
  __shared__ float s_ps[EE];
  __shared__ unsigned s_ct[EE];
  __shared__ float s_sq;

  const int tid = threadIdx.x;
  for (int i = tid; i < DIN*DCB; i += 256) {
    int d = i >> 9, c = i & (DIN-1);
    s_win[c*DCB + d] = win[i];
  }
  for (int i = tid; i < EE*KK*DCB; i += 256) {
    int e = i >> 10, r = i & 1023;
    s_cbn[e*(KK*DCB + 8) + r] = cbn[i];
  }
  for (int i = tid; i < EE*KK; i += 256) s_csq[i] = cbnsq[i];
  if (tid < DCB*DCB) s_w1[tid] = w1[tid];
  if (tid < EE*DCB)  s_w2[tid] = w2[tid];
  if (tid < DCB)     s_b1[tid] = b1[tid];
  if (tid < EE) { s_b2[tid] = b2[tid]; s_ps[tid] = 0.f; s_ct[tid] = 0u; }
  if (tid == 0) s_sq = 0.f;
  __syncthreads();

  const int tg = blockIdx.x*256 + tid;   // global token
  const int b  = tg / TT;
  const int t  = tg & (TT-1);
  const float* rp = resid + (size_t)b*DIN*TT + t;

  // ---- in_proj: z_e = W_in * residual_col + b_in (coalesced over t) ----
  float a[DCB];
  #pragma unroll
  for (int d = 0; d < DCB; ++d) a[d] = inb[d];
  #pragma unroll 8
  for (int c = 0; c < DIN; ++c) {
    float v = rp[(size_t)c*TT];
    #pragma unroll
    for (int d = 0; d < DCB; ++d) a[d] = fmaf(v, s_win[c*DCB + d], a[d]);
  }

  // ---- router: relu(z_e W1^T + b1) W2^T + b2; softmax; argmax ----
  float h[DCB];
  #pragma unroll
  for (int j = 0; j < DCB; ++j) {
    float hv = s_b1[j];
    #pragma unroll
    for (int d = 0; d < DCB; ++d) hv = fmaf(a[d], s_w1[j*DCB + d], hv);
    h[j] = fmaxf(hv, 0.f);
  }
  float lg[EE];
  float m = -3.402823466e+38f;
  int am = 0;
  #pragma unroll
  for (int e = 0; e < EE; ++e) {
    float lv = s_b2[e];
    #pragma unroll
    for (int j = 0; j < DCB; ++j) lv = fmaf(h[j], s_w2[e*DCB + j], lv);
    lg[e] = lv;
    if (lv > m) { m = lv; am = e; }   // first-max-wins == jnp.argmax
  }
  float psum = 0.f, p[EE];
  #pragma unroll
  for (int e = 0; e < EE; ++e) { p[e] = expf(lg[e] - m); psum += p[e]; }
  float pinv = 1.f / psum;
  #pragma unroll
  for (int e = 0; e < EE; ++e) atomicAdd(&s_ps[e], p[e]*pinv);
  atomicAdd(&s_ct[am], 1u);

  // ---- nearest codeword of the selected expert (normalized space) ----
  float n2 = 0.f;
  #pragma unroll
  for (int d = 0; d < DCB; ++d) n2 = fmaf(a[d], a[d], n2);
  float einv = 1.f / fmaxf(sqrtf(n2), 1e-12f);
  float enc[DCB];
  #pragma unroll
  for (int d = 0; d < DCB; ++d) enc[d] = a[d]*einv;

  const float* cb = &s_cbn[am*(KK*DCB + 8)];
  const float* cq = &s_csq[am*KK];
  float best = -3.402823466e+38f;
  int bk = 0;
  #pragma unroll 4
  for (int k = 0; k < KK; ++k) {
    float dot = 0.f;
    #pragma unroll
    for (int d = 0; d < DCB; ++d) dot = fmaf(enc[d], cb[k*DCB + d], dot);
    float sc = 2.f*dot - cq[k];       // argmax(-dist), ||enc||^2 const over k
    if (sc > best) { best = sc; bk = k; }
  }

  // ---- emit q (raw codebook), codes, latents, loss partials ----
  const float* cr = cbraw + (size_t)(am*KK + bk)*DCB;
  float qv[DCB];
  float sqp = 0.f;
  #pragma unroll
  for (int d = 0; d < DCB; ++d) {
    qv[d] = cr[d];
    float df = a[d] - qv[d];
    sqp = fmaf(df, df, sqp);
  }
  float4* qp = (float4*)(qout + (size_t)tg*DCB);
  qp[0] = make_float4(qv[0], qv[1], qv[2], qv[3]);
  qp[1] = make_float4(qv[4], qv[5], qv[6], qv[7]);

  codes[((size_t)b*NQ + qi)*TT + t] = (float)(am*KK + bk);

  float4* lp = (float4*)(lat + (((size_t)b*NQ + qi)*TT + t)*DCB);
  lp[0] = make_float4(a[0], a[1], a[2], a[3]);
  lp[1] = make_float4(a[4], a[5], a[6], a[7]);

  atomicAdd(&s_sq, sqp);
  __syncthreads();
  if (tid < EE) {
    atomicAdd(&acc[qi*EE + tid],      s_ps[tid]);          // probs sum
    atomicAdd(&acc[72 + qi*EE + tid], (float)s_ct[tid]);   // onehot count
  }
  if (tid == 0) atomicAdd(&acc[144 + qi], s_sq);
}

// ---------------------------------------------------------------------------
// Stage B (per quantizer): out = q @ w_out^T + out_b via V_WMMA_F32_16X16X4_F32,
// fused residual -= out. 32 tokens x 512 channels per block; 8 waves.
// A (16x4 f32):  lane L: M = L&15, VGPR0 K = 2*(L>>4),   VGPR1 K = 2*(L>>4)+1
// B (4x16 f32):  lane L: N = L&15, VGPR0 K = 2*(L>>4),   VGPR1 K = 2*(L>>4)+1
// D (16x16 f32): lane L: N = L&15, VGPR r: M = r + 8*(L>>4)
// ---------------------------------------------------------------------------
__global__ __launch_bounds__(256) void rvq_stageB(
    float* __restrict__ resid,         // [B][DIN][T], updated in place
    const float* __restrict__ qbuf,    // [BT][DCB]
    const float* __restrict__ wout,    // [DIN][DCB] weight-normed
    const float* __restrict__ outb)    // [DIN]
{
  __shared__ float s_w[DIN*DCB];       // [c][d]
  __shared__ float s_q[32*DCB];        // [tok][d]
  __shared__ float s_o[256*33];        // [cLocal][tok], pad 33 (bank-friendly)

  const int tid = threadIdx.x;
  const int tg0 = blockIdx.x*32;       // token base (T%32==0 -> never crosses b)
  for (int i = tid; i < DIN*DCB; i += 256) s_w[i] = wout[i];
  s_q[tid] = qbuf[(size_t)tg0*DCB + tid];
  __syncthreads();

  const int wave = tid >> 5;
  const int lane = tid & 31;
  const int mn = lane & 15;            // M for A / N for B,D
  const int kh = lane >> 4;            // K-half / M-half
  const int b  = tg0 / TT;
  const int t0 = tg0 & (TT-1);
  float* rbase = resid + (size_t)b*DIN*TT + t0;

  for (int half = 0; half < 2; ++half) {
    const int cbase = half*256;
    #pragma unroll
    for (int j = 0; j < 4; ++j) {
      int tile = wave + 8*j;           // 0..31 : 16 ctiles x 2 ttiles
      int ct = tile & 15;
      int tt = tile >> 4;
      int c0 = cbase + ct*16;
      const float* qrow = &s_q[(tt*16 + mn)*DCB];
      const float* wrow = &s_w[(c0 + mn)*DCB];
      v2f a0; a0[0] = qrow[2*kh];     a0[1] = qrow[2*kh + 1];
      v2f a1; a1[0] = qrow[4 + 2*kh]; a1[1] = qrow[5 + 2*kh];
      v2f b0; b0[0] = wrow[2*kh];     b0[1] = wrow[2*kh + 1];
      v2f b1; b1[0] = wrow[4 + 2*kh]; b1[1] = wrow[5 + 2*kh];
      v8f c = {0.f, 0.f, 0.f, 0.f, 0.f, 0.f, 0.f, 0.f};
      c = __builtin_amdgcn_wmma_f32_16x16x4_f32(false, a0, false, b0,
                                                (short)0, c, false, false);
      c = __builtin_amdgcn_wmma_f32_16x16x4_f32(false, a1, false, b1,
                                                (short)0, c, false, false);
      int cl = ct*16 + mn;             // local channel 0..255
      #pragma unroll
      for (int r = 0; r < 8; ++r)
        s_o[cl*33 + tt*16 + kh*8 + r] = c[r];
    }
    __syncthreads();
    // fused residual update for channels [cbase, cbase+256): coalesced over t
    #pragma unroll 4
    for (int it = 0; it < 32; ++it) {
      int idx = it*256 + tid;          // 0..8191
      int tt = idx & 31;
      int cl = idx >> 5;               // same for all 32 lanes -> scalar-ish
      int c  = cbase + cl;
      float o = s_o[cl*33 + tt] + outb[c];
      float* p = rbase + (size_t)c*TT + tt;
      *p = *p - o;
    }
    __syncthreads();
  }
}

// ---------------------------------------------------------------------------
// Final: z_q = z - residual_final (in place over the output z_q region).
// ---------------------------------------------------------------------------
__global__ __launch_bounds__(256) void rvq_final_zq(const float* __restrict__ z,
                                                    float* __restrict__ zq)
{
  size_t i = (size_t)blockIdx.x*256 + threadIdx.x;
  float4 a = ((const float4*)z)[i];
  float4 r = ((float4*)zq)[i];
  float4 o; o.x = a.x - r.x; o.y = a.y - r.y; o.z = a.z - r.z; o.w = a.w - r.w;
  ((float4*)zq)[i] = o;
}

__global__ void rvq_losses(const float* __restrict__ acc, float* __restrict__ loss)
{
  float commit = 0.f;
  for (int i = 0; i < NQ; ++i) commit += acc[144 + i];
  commit *= (1.0f / ((float)BT * (float)DCB));
  const float invBT = 1.0f / (float)BT;
  float aux = 0.f;
  for (int i = 0; i < NQ; ++i)
    for (int e = 0; e < EE; ++e)
      aux += (acc[72 + i*EE + e]*invBT) * (acc[i*EE + e]*invBT);
  loss[0] = commit;  // commitment_loss
  loss[1] = commit;  // codebook_loss (forward value identical)
  loss[2] = aux;     // aux_loss
}

// ---------------------------------------------------------------------------
extern "C" void kernel_launch(void* const* d_in, const int* in_sizes, int n_in,
                              void* d_out, int out_size, void* d_ws, size_t ws_size,
                              hipStream_t stream)
{
  const float* z     = (const float*)d_in[0];
  const float* in_v  = (const float*)d_in[1];
  const float* in_g  = (const float*)d_in[2];
  const float* in_b  = (const float*)d_in[3];
  const float* out_v = (const float*)d_in[4];
  const float* out_g = (const float*)d_in[5];
  const float* out_b = (const float*)d_in[6];
  const float* r_w1  = (const float*)d_in[7];
  const float* r_b1  = (const float*)d_in[8];
  const float* r_w2  = (const float*)d_in[9];
  const float* r_w2b = (const float*)d_in[10];
  const float* cbks  = (const float*)d_in[11];

  float* out = (float*)d_out;
  const size_t ZQ    = (size_t)BB*DIN*TT;      // 33554432
  const size_t CODES = (size_t)BB*NQ*TT;       // 589824
  const size_t LAT   = (size_t)BB*NQ*TT*DCB;   // 4718592
  float* o_zq    = out;                 // doubles as the residual buffer
  float* o_codes = out + ZQ;
  float* o_lat   = out + ZQ + CODES;
  float* o_loss  = out + ZQ + CODES + LAT;

  float* w        = (float*)d_ws;              // ~2.7 MB total
  float* ws_win   = w;                         // NQ*DCB*DIN  = 36864
  float* ws_wout  = ws_win  + NQ*DCB*DIN;      // NQ*DIN*DCB  = 36864
  float* ws_cbn   = ws_wout + NQ*DIN*DCB;      // NQ*EE*KK*DCB= 73728
  float* ws_cbnsq = ws_cbn  + NQ*EE*KK*DCB;    // NQ*EE*KK    = 9216
  float* ws_acc   = ws_cbnsq+ NQ*EE*KK;        // 160
  float* ws_q     = ws_acc  + 160;             // BT*DCB      = 524288

  const int prep_tasks = NQ*DCB + NQ*DIN + NQ*EE*KK + 160;  // 14056
  rvq_prep<<<(prep_tasks + 255)/256, 256, 0, stream>>>(
      in_v, in_g, out_v, out_g, cbks, ws_win, ws_wout, ws_cbn, ws_cbnsq, ws_acc);

  // residual := z (lives in the z_q output region)
  hipMemcpyAsync(o_zq, z, ZQ*sizeof(float), hipMemcpyDeviceToDevice, stream);

  for (int i = 0; i < NQ; ++i) {
    rvq_stageA<<<BT/256, 256, 0, stream>>>(
        o_zq,
        ws_win  + (size_t)i*DCB*DIN,
        in_b    + (size_t)i*DCB,
        r_w1    + (size_t)i*DCB*DCB,
        r_b1    + (size_t)i*DCB,
        r_w2    + (size_t)i*EE*DCB,
        r_w2b   + (size_t)i*EE,
        ws_cbn  + (size_t)i*EE*KK*DCB,
        ws_cbnsq+ (size_t)i*EE*KK,
        cbks    + (size_t)i*EE*KK*DCB,
        ws_q, o_codes, o_lat, ws_acc, i);
    rvq_stageB<<<BT/32, 256, 0, stream>>>(
        o_zq, ws_q,
        ws_wout + (size_t)i*DIN*DCB,
        out_b   + (size_t)i*DIN);
  }

  rvq_final_zq<<<(unsigned)(ZQ/4/256), 256, 0, stream>>>(z, o_zq);
  rvq_losses<<<1, 1, 0, stream>>>(ws_acc, o_loss);
}